// LaneSetLoss_20856361190079
// MI455X (gfx1250) — compile-verified
//
#include <hip/hip_runtime.h>
#include <math.h>

// ---------------------------------------------------------------------------
// LaneSetLoss for MI455X (gfx1250, wave32).
// B=2, Mpred=40, Ngt=24, P=24 points/curve, NRS=64, grid 72x128=9216.
// Soft-mask rasterizer uses V_WMMA_F32_16X16X4_F32 (exact f32, K=2 padded to 4):
//   segments -> M, grid points -> N; C-tile min over segments is 8 in-lane
//   VGPRs + one half-wave shuffle. Inner loop is branchless (invalid-segment
//   penalty folded into |a|^2) and division-free (precomputed 1/den).
// ---------------------------------------------------------------------------

typedef float v2f __attribute__((ext_vector_type(2)));
typedef float v8f __attribute__((ext_vector_type(8)));

#define HGRID 72
#define WGRID 128
#define NGRID (HGRID * WGRID)   // 9216
#define NPTS 24
#define NSEG 23
#define NRS 64
#define MPRED 40
#define NGT 24
#define NB 2
#define THICK 0.03f
#define SHARP 80.0f
#define BIG2V 1.0e18f

__device__ __forceinline__ v8f wmma_k4(v2f a, v2f b, v8f c) {
  // D = A(16x4 f32) * B(4x16 f32) + C(16x16 f32)
  return __builtin_amdgcn_wmma_f32_16x16x4_f32(
      /*neg_a=*/false, a, /*neg_b=*/false, b,
      /*c_mod=*/(short)0, c, /*reuse_a=*/false, /*reuse_b=*/false);
}

__device__ __forceinline__ float smooth_l1(float x, float y) {
  float d = fabsf(x - y);
  return d < 1.0f ? 0.5f * d * d : d - 0.5f;
}

// ---------------------------------------------------------------------------
// Kernel 1: arc-length resampling (1 wave per curve) + gt mask prep.
// grid = NB * 64 blocks of 32 threads. c < 40 -> pred curve, else gt curve.
// ---------------------------------------------------------------------------
__global__ __launch_bounds__(32) void k_resample(
    const float* __restrict__ pred_pts, const float* __restrict__ gt_pts,
    const float* __restrict__ gt_vis,
    float* __restrict__ pred_rs, float* __restrict__ gt_rs,
    float* __restrict__ gt_pm_out) {
  int bid = blockIdx.x;
  int c = bid % 64;
  int b = bid / 64;
  bool isPred = c < MPRED;
  int curve = isPred ? c : c - MPRED;
  const float* pts = isPred ? pred_pts + ((size_t)(b * MPRED + curve)) * NPTS * 2
                            : gt_pts + ((size_t)(b * NGT + curve)) * NPTS * 2;

  __shared__ float s_px[NPTS], s_py[NPTS], s_pm[NPTS];
  __shared__ float s_seg[NSEG];
  __shared__ float s_cum[NPTS];

  int lane = threadIdx.x;
  if (lane < NPTS) {
    s_px[lane] = pts[lane * 2 + 0];
    s_py[lane] = pts[lane * 2 + 1];
    s_pm[lane] = 1.0f;
  }
  __syncthreads();
  if (!isPred && lane == 0) {
    const float* vis = gt_vis + (size_t)(b * NGT + curve) * NPTS;
    int cnt = 0;
    float raw[NPTS];
    for (int k = 0; k < NPTS; ++k) {
      raw[k] = vis[k] > 0.5f ? 1.0f : 0.0f;
      cnt += (raw[k] > 0.5f) ? 1 : 0;
    }
    for (int k = 0; k < NPTS; ++k) {
      float v = (cnt < 2) ? 1.0f : raw[k];
      s_pm[k] = v;
      gt_pm_out[(size_t)(b * NGT + curve) * NPTS + k] = v;
    }
  }
  __syncthreads();
  if (lane < NSEG) {
    float dx = s_px[lane + 1] - s_px[lane];
    float dy = s_py[lane + 1] - s_py[lane];
    s_seg[lane] = sqrtf(dx * dx + dy * dy) * (s_pm[lane] * s_pm[lane + 1]);
  }
  __syncthreads();
  if (lane == 0) {
    float acc = 0.0f;
    s_cum[0] = 0.0f;
    for (int s = 0; s < NSEG; ++s) { acc += s_seg[s]; s_cum[s + 1] = acc; }
  }
  __syncthreads();
  float total = s_cum[NSEG];
  float* dst = isPred ? pred_rs + (size_t)(b * MPRED + curve) * NRS * 2
                      : gt_rs + (size_t)(b * NGT + curve) * NRS * 2;
  for (int k = lane; k < NRS; k += 32) {
    float t = ((float)k * (1.0f / 63.0f)) * total;
    int idx = -1;
    for (int s = 0; s < NPTS; ++s) idx += (s_cum[s] <= t) ? 1 : 0;  // searchsorted right - 1
    idx = idx < 0 ? 0 : (idx > NSEG - 1 ? NSEG - 1 : idx);
    float lt = s_cum[idx], rt = s_cum[idx + 1];
    float alpha = (t - lt) / fmaxf(rt - lt, 1e-8f);
    float ox = s_px[idx] + alpha * (s_px[idx + 1] - s_px[idx]);
    float oy = s_py[idx] + alpha * (s_py[idx + 1] - s_py[idx]);
    if (total < 1e-8f) { ox = s_px[0]; oy = s_py[0]; }
    dst[k * 2 + 0] = ox;
    dst[k * 2 + 1] = oy;
  }
}

// ---------------------------------------------------------------------------
// Kernel 2: pairwise sym/tangent/curvature partial cost.
// grid = NB*MPRED*NGT blocks of 64 threads (thread k == resampled point k).
// Per-segment (a, ab, 1/den) precomputed in LDS -> division-free inner loop.
// ---------------------------------------------------------------------------
__device__ __forceinline__ float block_reduce64(float v, float* red, int tid) {
  red[tid] = v;
  __syncthreads();
  for (int s = 32; s > 0; s >>= 1) {
    if (tid < s) red[tid] += red[tid + s];
    __syncthreads();
  }
  float r = red[0];
  __syncthreads();
  return r;
}

__global__ __launch_bounds__(64) void k_pair(
    const float* __restrict__ pred_rs, const float* __restrict__ gt_rs,
    float* __restrict__ partial) {
  int bid = blockIdx.x;
  int j = bid % NGT;
  int t2 = bid / NGT;
  int i = t2 % MPRED;
  int b = t2 / MPRED;

  __shared__ float Px[NRS], Py[NRS], Gx[NRS], Gy[NRS];
  __shared__ float PSax[NRS - 1], PSay[NRS - 1], PSbx[NRS - 1], PSby[NRS - 1], PSr[NRS - 1];
  __shared__ float GSax[NRS - 1], GSay[NRS - 1], GSbx[NRS - 1], GSby[NRS - 1], GSr[NRS - 1];
  __shared__ float red[64];
  int tid = threadIdx.x;
  const float* P = pred_rs + (size_t)(b * MPRED + i) * NRS * 2;
  const float* G = gt_rs + (size_t)(b * NGT + j) * NRS * 2;
  Px[tid] = P[tid * 2 + 0];
  Py[tid] = P[tid * 2 + 1];
  Gx[tid] = G[tid * 2 + 0];
  Gy[tid] = G[tid * 2 + 1];
  __syncthreads();
  if (tid < NRS - 1) {
    float ax = Px[tid], ay = Py[tid];
    float bx = Px[tid + 1] - ax, by = Py[tid + 1] - ay;
    PSax[tid] = ax; PSay[tid] = ay; PSbx[tid] = bx; PSby[tid] = by;
    PSr[tid] = 1.0f / fmaxf(bx * bx + by * by, 1e-8f);
    ax = Gx[tid]; ay = Gy[tid];
    bx = Gx[tid + 1] - ax; by = Gy[tid + 1] - ay;
    GSax[tid] = ax; GSay[tid] = ay; GSbx[tid] = bx; GSby[tid] = by;
    GSr[tid] = 1.0f / fmaxf(bx * bx + by * by, 1e-8f);
  }
  __syncthreads();

  // point -> polyline (min over 63 segments, squared then sqrt)
  float dpg, dgp;
  {
    float px = Px[tid], py = Py[tid];
    float best = 3.4e38f;
    for (int s = 0; s < NRS - 1; ++s) {
      float dx0 = px - GSax[s], dy0 = py - GSay[s];
      float bx = GSbx[s], by = GSby[s];
      float t = fminf(fmaxf((dx0 * bx + dy0 * by) * GSr[s], 0.0f), 1.0f);
      float dx = dx0 - t * bx, dy = dy0 - t * by;
      best = fminf(best, dx * dx + dy * dy);
    }
    dpg = sqrtf(best);
  }
  {
    float px = Gx[tid], py = Gy[tid];
    float best = 3.4e38f;
    for (int s = 0; s < NRS - 1; ++s) {
      float dx0 = px - PSax[s], dy0 = py - PSay[s];
      float bx = PSbx[s], by = PSby[s];
      float t = fminf(fmaxf((dx0 * bx + dy0 * by) * PSr[s], 0.0f), 1.0f);
      float dx = dx0 - t * bx, dy = dy0 - t * by;
      best = fminf(best, dx * dx + dy * dy);
    }
    dgp = sqrtf(best);
  }

  // tangents (central differences, endpoint one-sided)
  float fx, fy, gx2, gy2;
  if (tid == 0)            { fx = Px[1] - Px[0];   fy = Py[1] - Py[0]; }
  else if (tid == NRS - 1) { fx = Px[63] - Px[62]; fy = Py[63] - Py[62]; }
  else                     { fx = Px[tid + 1] - Px[tid - 1]; fy = Py[tid + 1] - Py[tid - 1]; }
  if (tid == 0)            { gx2 = Gx[1] - Gx[0];   gy2 = Gy[1] - Gy[0]; }
  else if (tid == NRS - 1) { gx2 = Gx[63] - Gx[62]; gy2 = Gy[63] - Gy[62]; }
  else                     { gx2 = Gx[tid + 1] - Gx[tid - 1]; gy2 = Gy[tid + 1] - Gy[tid - 1]; }
  float nf = fmaxf(sqrtf(fx * fx + fy * fy), 1e-8f);
  float ng = fmaxf(sqrtf(gx2 * gx2 + gy2 * gy2), 1e-8f);
  float tansum = fabsf((fx / nf) * (gx2 / ng) + (fy / nf) * (gy2 / ng));

  // curvature smooth-L1 (62 second differences, 2 comps)
  float curv = 0.0f;
  if (tid < NRS - 2) {
    float psx = Px[tid + 2] - 2.0f * Px[tid + 1] + Px[tid];
    float psy = Py[tid + 2] - 2.0f * Py[tid + 1] + Py[tid];
    float gsx = Gx[tid + 2] - 2.0f * Gx[tid + 1] + Gx[tid];
    float gsy = Gy[tid + 2] - 2.0f * Gy[tid + 1] + Gy[tid];
    curv = smooth_l1(psx, gsx) + smooth_l1(psy, gsy);
  }

  float Sdpg = block_reduce64(dpg, red, tid);
  float Sdgp = block_reduce64(dgp, red, tid);
  float Stan = block_reduce64(tansum, red, tid);
  float Scur = block_reduce64(curv, red, tid);

  if (tid == 0) {
    float sym = 0.5f * (Sdpg * (1.0f / NRS) + Sdgp * (1.0f / NRS));
    float tan = 1.0f - Stan * (1.0f / NRS);
    float cur = Scur * (1.0f / ((NRS - 2) * 2));
    partial[(size_t)(b * MPRED + i) * NGT + j] = 5.0f * sym + 1.0f * tan + 0.5f * cur;
  }
}

// ---------------------------------------------------------------------------
// Kernel 3: soft-mask rasterizer with V_WMMA_F32_16X16X4_F32.
// 1 wave per block; each wave handles one curve x 64 grid points.
// grid = NB * 64 * (NGRID/64) = 18432 blocks of 32 threads.
// Per-element math is branchless & division-free:
//   t    = clamp((x.ab - a.ab) * (1/den), 0, 1)
//   d^2  = (|x|^2 - 2 x.a + |a|^2 + penalty) - 2 t u + t^2 den
// where penalty = 1e18 for invalid/padded segments (folded into |a|^2).
// ---------------------------------------------------------------------------
__global__ __launch_bounds__(32) void k_mask(
    const float* __restrict__ pred_pts, const float* __restrict__ gt_pts,
    const float* __restrict__ gt_pm,
    float* __restrict__ pmask, float* __restrict__ gmask) {
  int bid = blockIdx.x;
  int gtile = bid % (NGRID / 64);
  int t2 = bid / (NGRID / 64);
  int c = t2 % 64;
  int b = t2 / 64;
  bool isPred = c < MPRED;
  int curve = isPred ? c : c - MPRED;
  const float* pts = isPred ? pred_pts + (size_t)(b * MPRED + curve) * NPTS * 2
                            : gt_pts + (size_t)(b * NGT + curve) * NPTS * 2;

  __shared__ float s_px[NPTS], s_py[NPTS], s_pm[NPTS];
  __shared__ float s_ax[32], s_ay[32], s_abx[32], s_aby[32];
  __shared__ float s_den[32], s_rden[32], s_adab[32], s_a2p[32];
  __shared__ int s_novalid;

  int lane = threadIdx.x;
  if (lane < NPTS) {
    s_px[lane] = pts[lane * 2 + 0];
    s_py[lane] = pts[lane * 2 + 1];
    s_pm[lane] = isPred ? 1.0f : gt_pm[(size_t)(b * NGT + curve) * NPTS + lane];
  }
  __syncthreads();
  if (lane < NSEG) {
    float ax = s_px[lane], ay = s_py[lane];
    float abx = s_px[lane + 1] - ax, aby = s_py[lane + 1] - ay;
    float den = fmaxf(abx * abx + aby * aby, 1e-8f);
    bool valid = (s_pm[lane] * s_pm[lane + 1] > 0.5f);
    s_ax[lane] = ax; s_ay[lane] = ay; s_abx[lane] = abx; s_aby[lane] = aby;
    s_den[lane] = den;
    s_rden[lane] = 1.0f / den;
    s_adab[lane] = ax * abx + ay * aby;
    s_a2p[lane] = ax * ax + ay * ay + (valid ? 0.0f : BIG2V);  // penalty folded in
  } else {
    s_ax[lane] = 0.0f; s_ay[lane] = 0.0f; s_abx[lane] = 0.0f; s_aby[lane] = 0.0f;
    s_den[lane] = 1e-8f; s_rden[lane] = 1e8f; s_adab[lane] = 0.0f;
    s_a2p[lane] = BIG2V;  // padded segments always invalid
  }
  __syncthreads();
  if (lane == 0) {
    int any = 0;
    for (int s = 0; s < NSEG; ++s) any |= (s_pm[s] * s_pm[s + 1] > 0.5f) ? 1 : 0;
    s_novalid = !any;
  }
  __syncthreads();

  // A operands: segments along M. Lanes 0-15 carry K=0,1 (=ab.x,ab.y / a.x,a.y);
  // lanes 16-31 carry K=2,3 padding -> zero.
  v2f A_ab0 = {0.0f, 0.0f}, A_a0 = {0.0f, 0.0f};
  v2f A_ab1 = {0.0f, 0.0f}, A_a1 = {0.0f, 0.0f};
  if (lane < 16) {
    A_ab0 = {s_abx[lane], s_aby[lane]};
    A_a0  = {s_ax[lane],  s_ay[lane]};
    A_ab1 = {s_abx[lane + 16], s_aby[lane + 16]};
    A_a1  = {s_ax[lane + 16],  s_ay[lane + 16]};
  }

  int halfsel = (lane >> 4) << 3;  // 0 for lanes 0-15 (M=r), 8 for lanes 16-31 (M=r+8)
  int col = lane & 15;
  int gbase = gtile * 64;
  float* dst = isPred ? pmask + (size_t)(b * MPRED + curve) * NGRID
                      : gmask + (size_t)(b * NGT + curve) * NGRID;

#pragma unroll
  for (int n = 0; n < 4; ++n) {
    int gp = gbase + n * 16 + col;
    int w = gp & (WGRID - 1);
    int h = gp >> 7;
    float x = (float)w * (1.0f / (WGRID - 1));
    float y = (float)h * (1.0f / (HGRID - 1));
    v2f Bv = {0.0f, 0.0f};
    if (lane < 16) Bv = {x, y};  // K=0,1 rows; lanes 16-31 = K=2,3 pad -> zero
    float x2 = x * x + y * y;

    v8f zero = {};
    v8f c0 = wmma_k4(A_ab0, Bv, zero);  // (seg 0..15)  x . ab
    v8f d0 = wmma_k4(A_a0,  Bv, zero);  // (seg 0..15)  x . a
    v8f c1 = wmma_k4(A_ab1, Bv, zero);  // (seg 16..31) x . ab
    v8f d1 = wmma_k4(A_a1,  Bv, zero);  // (seg 16..31) x . a

    float mind2 = BIG2V;
#pragma unroll
    for (int r = 0; r < 8; ++r) {
      {
        int seg = r + halfsel;
        float u = c0[r] - s_adab[seg];
        float t = fminf(fmaxf(u * s_rden[seg], 0.0f), 1.0f);
        float dd = x2 - 2.0f * d0[r] + s_a2p[seg] - 2.0f * t * u + t * t * s_den[seg];
        mind2 = fminf(mind2, fmaxf(dd, 0.0f));
      }
      {
        int seg = 16 + r + halfsel;
        float u = c1[r] - s_adab[seg];
        float t = fminf(fmaxf(u * s_rden[seg], 0.0f), 1.0f);
        float dd = x2 - 2.0f * d1[r] + s_a2p[seg] - 2.0f * t * u + t * t * s_den[seg];
        mind2 = fminf(mind2, fmaxf(dd, 0.0f));
      }
    }
    // combine the two lane-halves (each grid point lives in lane L and L+16)
    mind2 = fminf(mind2, __shfl_xor(mind2, 16));

    if (s_novalid) {
      // fallback: min point distance over visible points
      mind2 = BIG2V;
      for (int k = 0; k < NPTS; ++k) {
        if (s_pm[k] > 0.5f) {
          float dx = x - s_px[k], dy = y - s_py[k];
          mind2 = fminf(mind2, dx * dx + dy * dy);
        }
      }
    }

    float md = sqrtf(mind2);
    float mask = 1.0f / (1.0f + expf((md - THICK) * SHARP));
    if (lane < 16) dst[gp] = mask;
  }
}

// ---------------------------------------------------------------------------
// Kernel 4: pairwise soft-IoU (exact min/max) + final combine + exist gate.
// grid = NB*MPRED*NGT blocks of 256 threads; float4 (b128) streaming.
// ---------------------------------------------------------------------------
__global__ __launch_bounds__(256) void k_iou(
    const float* __restrict__ pmask, const float* __restrict__ gmask,
    const float* __restrict__ partial, const float* __restrict__ gt_exist,
    float* __restrict__ out) {
  int bid = blockIdx.x;
  int j = bid % NGT;
  int t2 = bid / NGT;
  int i = t2 % MPRED;
  int b = t2 / MPRED;

  const float4* p = (const float4*)(pmask + (size_t)(b * MPRED + i) * NGRID);
  const float4* g = (const float4*)(gmask + (size_t)(b * NGT + j) * NGRID);

  float inter = 0.0f, uni = 0.0f;
  for (int x = threadIdx.x; x < NGRID / 4; x += 256) {
    float4 a = p[x];
    float4 c = g[x];
    inter += fminf(a.x, c.x) + fminf(a.y, c.y) + fminf(a.z, c.z) + fminf(a.w, c.w);
    uni   += fmaxf(a.x, c.x) + fmaxf(a.y, c.y) + fmaxf(a.z, c.z) + fmaxf(a.w, c.w);
  }
  __shared__ float r1[256], r2[256];
  int tid = threadIdx.x;
  r1[tid] = inter;
  r2[tid] = uni;
  __syncthreads();
  for (int s = 128; s > 0; s >>= 1) {
    if (tid < s) { r1[tid] += r1[tid + s]; r2[tid] += r2[tid + s]; }
    __syncthreads();
  }
  if (tid == 0) {
    float overlap = 1.0f - r1[0] / (r2[0] + 1e-8f);
    float cost = partial[(size_t)(b * MPRED + i) * NGT + j] + 2.0f * overlap;
    float e = gt_exist[b * NGT + j] > 0.5f ? 1.0f : 0.0f;
    out[(size_t)(b * MPRED + i) * NGT + j] = cost * e;
  }
}

// ---------------------------------------------------------------------------
// Launch
// ---------------------------------------------------------------------------
extern "C" void kernel_launch(void* const* d_in, const int* in_sizes, int n_in,
                              void* d_out, int out_size, void* d_ws, size_t ws_size,
                              hipStream_t stream) {
  (void)in_sizes; (void)n_in; (void)out_size; (void)ws_size;
  const float* pred_pts = (const float*)d_in[0];  // (2,40,24,2)
  const float* gt_pts   = (const float*)d_in[1];  // (2,24,24,2)
  const float* gt_vis   = (const float*)d_in[2];  // (2,24,24)
  const float* gt_exist = (const float*)d_in[3];  // (2,24)
  float* out = (float*)d_out;                     // (2,40,24)

  float* ws = (float*)d_ws;
  float* pred_rs = ws + 0;        // 2*40*64*2 = 10240
  float* gt_rs   = ws + 10240;    // 2*24*64*2 = 6144
  float* gt_pm   = ws + 16384;    // 2*24*24   = 1152
  float* partial = ws + 17536;    // 2*40*24   = 1920
  float* pmask   = ws + 19456;    // 2*40*9216 = 737280 (16B-aligned offset)
  float* gmask   = ws + 756736;   // 2*24*9216 = 442368 (16B-aligned offset)

  k_resample<<<NB * 64, 32, 0, stream>>>(pred_pts, gt_pts, gt_vis,
                                         pred_rs, gt_rs, gt_pm);
  k_pair<<<NB * MPRED * NGT, 64, 0, stream>>>(pred_rs, gt_rs, partial);
  k_mask<<<NB * 64 * (NGRID / 64), 32, 0, stream>>>(pred_pts, gt_pts, gt_pm,
                                                    pmask, gmask);
  k_iou<<<NB * MPRED * NGT, 256, 0, stream>>>(pmask, gmask, partial, gt_exist, out);
}